// DLP_Loss_24696061952372
// MI455X (gfx1250) — compile-verified
//
#include <hip/hip_runtime.h>

#define NROWS 2048
#define NCH   256
#define NCLS  8
#define KSEL  20
#define LAMW  50.0f
#define EPSW  1e-6f

typedef float v2f __attribute__((ext_vector_type(2)));
typedef float v8f __attribute__((ext_vector_type(8)));

// ---------------- per-row sum of squares: S[r] = sum_c X[r,c]^2 ----------------
__global__ __launch_bounds__(256) void rowsumsq_kernel(const float* __restrict__ X,
                                                       float* __restrict__ S) {
    __shared__ float red[8];
    const int t = threadIdx.x, r = blockIdx.x;
    const int w = t >> 5, lane = t & 31;
    const float v = X[r * NCH + t];
    float sq = v * v;
    sq += __shfl_xor(sq, 16, 32);
    sq += __shfl_xor(sq, 8, 32);
    sq += __shfl_xor(sq, 4, 32);
    sq += __shfl_xor(sq, 2, 32);
    sq += __shfl_xor(sq, 1, 32);
    if (lane == 0) red[w] = sq;
    __syncthreads();
    if (t == 0) {
        float s = 0.0f;
        for (int q = 0; q < 8; ++q) s += red[q];
        S[r] = s;
    }
}

// ---------------- per-row cross entropy ----------------
__global__ __launch_bounds__(256) void ce_kernel(const float* __restrict__ scores,
                                                 const int* __restrict__ tgt,
                                                 float* __restrict__ ce) {
    const int r = blockIdx.x * blockDim.x + threadIdx.x;
    if (r >= NROWS) return;
    const float* s = scores + r * NCLS;
    float mx = s[0];
#pragma unroll
    for (int c = 1; c < NCLS; ++c) mx = fmaxf(mx, s[c]);
    float sum = 0.0f;
#pragma unroll
    for (int c = 0; c < NCLS; ++c) sum += expf(s[c] - mx);
    const float lse = mx + logf(sum);
    ce[r] = lse - s[tgt[r]];
}

// ---------------- main: L1 distances, same-label top-K, WMMA neighbor term ----------------
__global__ __launch_bounds__(256) void dlp_main_kernel(const float* __restrict__ X,
                                                       const int* __restrict__ tgt,
                                                       const float* __restrict__ S,
                                                       float* __restrict__ rowLoss) {
    __shared__ float xis[NCH];
    __shared__ float Ds[NROWS];
    __shared__ float redV[8];
    __shared__ int   redI[8];
    __shared__ int   nk[32];
    __shared__ int   mSh;

    const int i    = blockIdx.x;
    const int t    = threadIdx.x;
    const int w    = t >> 5;
    const int lane = t & 31;
    const int ti   = tgt[i];
    const float FINF = __builtin_inff();

    // Phase 0: stage row i in LDS via async global->LDS copy (ASYNCcnt path).
    // LDS byte address = low 32 bits of the flat shared pointer (ISA 10.2).
    {
        unsigned lds_addr = (unsigned)(uintptr_t)(&xis[t]);
        const float* gp = X + i * NCH + t;
        asm volatile("global_load_async_to_lds_b32 %0, %1, off"
                     :: "v"(lds_addr), "v"(gp) : "memory");
        asm volatile("s_wait_asynccnt 0" ::: "memory");
    }
    __syncthreads();

    // Phase 1: D[i,j] = sum_c |x_i[c] + eps - x_j[c]|  (wave per j-stripe,
    // lane owns an 8-channel slice -> coalesced b128 row reads)
    const float4* xi4 = reinterpret_cast<const float4*>(&xis[lane * 8]);
    float4 xa = xi4[0], xb = xi4[1];
    xa.x += EPSW; xa.y += EPSW; xa.z += EPSW; xa.w += EPSW;
    xb.x += EPSW; xb.y += EPSW; xb.z += EPSW; xb.w += EPSW;

    for (int jj = w; jj < NROWS; jj += 8) {
        const float4* rp = reinterpret_cast<const float4*>(X + jj * NCH + lane * 8);
        const float4 v0 = rp[0];
        const float4 v1 = rp[1];
        if (jj + 8 < NROWS)
            __builtin_prefetch(X + (jj + 8) * NCH + lane * 8, 0, 3);  // global_prefetch_b8
        float acc = fabsf(xa.x - v0.x) + fabsf(xa.y - v0.y) +
                    fabsf(xa.z - v0.z) + fabsf(xa.w - v0.w) +
                    fabsf(xb.x - v1.x) + fabsf(xb.y - v1.y) +
                    fabsf(xb.z - v1.z) + fabsf(xb.w - v1.w);
        acc += __shfl_xor(acc, 16, 32);
        acc += __shfl_xor(acc, 8, 32);
        acc += __shfl_xor(acc, 4, 32);
        acc += __shfl_xor(acc, 2, 32);
        acc += __shfl_xor(acc, 1, 32);
        if (lane == 0) Ds[jj] = acc;
    }
    __syncthreads();

    // Phase 1.5: mask (same label, not self) + candidate count.
    // j = t + 256*q: conflict-free LDS banks + coalesced tgt reads.
    int cnt = 0;
#pragma unroll
    for (int q = 0; q < 8; ++q) {
        const int j = t + 256 * q;
        if (tgt[j] != ti || j == i) Ds[j] = FINF;
        else ++cnt;
    }
    cnt += __shfl_xor(cnt, 16, 32);
    cnt += __shfl_xor(cnt, 8, 32);
    cnt += __shfl_xor(cnt, 4, 32);
    cnt += __shfl_xor(cnt, 2, 32);
    cnt += __shfl_xor(cnt, 1, 32);
    if (lane == 0) redI[w] = cnt;
    __syncthreads();
    if (t == 0) {
        int c2 = 0;
        for (int q = 0; q < 8; ++q) c2 += redI[q];
        mSh = (c2 < KSEL) ? c2 : KSEL;
    }
    __syncthreads();

    // Phase 2: K argmin passes; wave-level (val,idx) shfl reduction,
    // only 2 barriers per pass. Index tie-break == top_k ordering.
    for (int it = 0; it < KSEL; ++it) {
        float bv = FINF;
        int   bi = 0;
#pragma unroll
        for (int q = 0; q < 8; ++q) {
            const int j = t + 256 * q;       // ascending within thread
            const float v = Ds[j];
            if (v < bv) { bv = v; bi = j; }
        }
#pragma unroll
        for (int m = 16; m > 0; m >>= 1) {
            const float ov = __shfl_xor(bv, m, 32);
            const int   oi = __shfl_xor(bi, m, 32);
            if (ov < bv || (ov == bv && oi < bi)) { bv = ov; bi = oi; }
        }
        if (lane == 0) { redV[w] = bv; redI[w] = bi; }
        __syncthreads();
        if (t == 0) {
            float v0 = redV[0];
            int   i0 = redI[0];
            for (int q = 1; q < 8; ++q) {
                if (redV[q] < v0 || (redV[q] == v0 && redI[q] < i0)) {
                    v0 = redV[q]; i0 = redI[q];
                }
            }
            if (v0 < FINF) { nk[it] = i0; Ds[i0] = FINF; }
            else           { nk[it] = 0; }
        }
        __syncthreads();
    }
    if (t == 0) {
        for (int q = KSEL; q < 32; ++q) nk[q] = 0;
    }
    __syncthreads();

    // Phase 3 (wave 0, EXEC all-ones): Gram values G_k = <x_i, y_k> via
    // V_WMMA_F32_16X16X4_F32. A = 16 neighbor rows (M=lane&15, K pair via
    // lane half), B = x_i broadcast into all 16 columns; read column N=0.
    if (t < 32) {
        const int l15  = lane & 15;
        const int coff = (lane >> 4) << 1;           // K offset 0 or 2
        const int r1 = nk[l15];                      // chain 1: k = 0..15
        const int r2 = (l15 < 4) ? nk[16 + l15] : 0; // chain 2: k = 16..19 (+pad)
        const float* p1 = X + r1 * NCH + coff;
        const float* p2 = X + r2 * NCH + coff;

        v8f acc1 = {0.f, 0.f, 0.f, 0.f, 0.f, 0.f, 0.f, 0.f};
        v8f acc2 = {0.f, 0.f, 0.f, 0.f, 0.f, 0.f, 0.f, 0.f};
        for (int c = 0; c < NCH; c += 4) {
            const v2f A1 = *reinterpret_cast<const v2f*>(p1 + c);
            const v2f A2 = *reinterpret_cast<const v2f*>(p2 + c);
            const v2f Bv = *reinterpret_cast<const v2f*>(&xis[c + coff]);
            acc1 = __builtin_amdgcn_wmma_f32_16x16x4_f32(false, A1, false, Bv,
                                                         (short)0, acc1, false, false);
            acc2 = __builtin_amdgcn_wmma_f32_16x16x4_f32(false, A2, false, Bv,
                                                         (short)0, acc2, false, false);
        }

        const int   mlim = mSh;
        const float invm = 1.0f / (float)((mlim > 0) ? mlim : 1);
        const float Si   = S[i];
        float part = 0.0f;
        if (lane == 0) {
            // column 0, M=0..7 -> k=0..7 ; chain2 M=0..3 -> k=16..19
            for (int q = 0; q < 8; ++q)
                if (q < mlim) part += Si - 2.0f * invm * acc1[q] + S[nk[q]] * invm * invm;
            for (int q = 0; q < 4; ++q)
                if (16 + q < mlim) part += Si - 2.0f * invm * acc2[q] + S[nk[16 + q]] * invm * invm;
        } else if (lane == 16) {
            // column 0, M=8..15 -> k=8..15
            for (int q = 0; q < 8; ++q)
                if (8 + q < mlim) part += Si - 2.0f * invm * acc1[q] + S[nk[8 + q]] * invm * invm;
        }
        const float tot = __shfl(part, 0, 32) + __shfl(part, 16, 32);
        if (lane == 0) rowLoss[i] = tot;
    }
}

// ---------------- deterministic fixed-order finalize ----------------
__global__ __launch_bounds__(256) void finalize_kernel(const float* __restrict__ ce,
                                                       const float* __restrict__ rl,
                                                       float* __restrict__ out) {
    __shared__ float r1[256];
    __shared__ float r2[256];
    const int t = threadIdx.x;
    float a = 0.0f, b = 0.0f;
    for (int j = t; j < NROWS; j += 256) { a += ce[j]; b += rl[j]; }
    r1[t] = a; r2[t] = b;
    __syncthreads();
    for (int s = 128; s > 0; s >>= 1) {
        if (t < s) { r1[t] += r1[t + s]; r2[t] += r2[t + s]; }
        __syncthreads();
    }
    if (t == 0) out[0] = r1[0] / (float)NROWS + LAMW * 0.5f * r2[0];
}

extern "C" void kernel_launch(void* const* d_in, const int* in_sizes, int n_in,
                              void* d_out, int out_size, void* d_ws, size_t ws_size,
                              hipStream_t stream) {
    (void)in_sizes; (void)n_in; (void)out_size; (void)ws_size;
    const float* feature = (const float*)d_in[0];
    const float* scores  = (const float*)d_in[1];
    const int*   target  = (const int*)d_in[2];
    float* out = (float*)d_out;

    float* S  = (float*)d_ws;      // [NROWS]
    float* ce = S + NROWS;         // [NROWS]
    float* rl = ce + NROWS;        // [NROWS]

    rowsumsq_kernel<<<NROWS, 256, 0, stream>>>(feature, S);
    ce_kernel<<<NROWS / 256, 256, 0, stream>>>(scores, target, ce);
    dlp_main_kernel<<<NROWS, 256, 0, stream>>>(feature, target, S, rl);
    finalize_kernel<<<1, 256, 0, stream>>>(ce, rl, out);
}